// ClusterSeedClsOld_32555852103939
// MI455X (gfx1250) — compile-verified
//
#include <hip/hip_runtime.h>

// ---------------- problem constants ----------------
#define N_PIX   (1u << 21)          // 1024 * 2048
#define NWORD   (N_PIX / 32u)       // 65536 packed 32-bit mask words
#define W_      2048
#define NBLK    512
#define NTHR    256
#define NWAVE   ((NBLK * NTHR) / 32)   // 4096 waves; NWORD/NWAVE == 16 exactly
#define MAXIT   (1 << 22)

#define THRESHOLD      0.5f
#define MIN_PIXEL      160
#define MIN_INST_PIXEL 160
#define INST_RATIO     0.5f
#define DIST_THRESH    0.5f
#define MIN_SEED       0.5f

typedef __attribute__((ext_vector_type(2))) float v2f;
typedef __attribute__((ext_vector_type(8))) float v8f;
typedef __attribute__((ext_vector_type(2))) int   v2i;

// CDNA5 async global->LDS path (guarded: falls back to direct loads if absent)
#if defined(__has_builtin)
#if __has_builtin(__builtin_amdgcn_global_load_async_to_lds_b64) && \
    __has_builtin(__builtin_amdgcn_s_wait_asynccnt)
#define HAVE_ASYNC_LDS 1
#endif
#endif
#ifndef HAVE_ASYNC_LDS
#define HAVE_ASYNC_LDS 0
#endif

#if HAVE_ASYNC_LDS
// b64 variant moves one v2i (8 bytes) per lane: global -> LDS, tracked by ASYNCcnt.
__device__ __forceinline__ void async_copy_b64(const void* g, void* l) {
  __builtin_amdgcn_global_load_async_to_lds_b64(
      (__attribute__((address_space(1))) v2i*)g,
      (__attribute__((address_space(3))) v2i*)l,
      0, 0);
}
#endif

struct Ctl {
  unsigned barCounter, barGen;
  unsigned finished;
  unsigned long long argmaxPacked;   // (scoreBits<<32) | ~idx  (score>=0 => monotone)
  unsigned unclCnt;
  unsigned psum, overlap;
  unsigned count;                    // next instance id (starts at 1)
  unsigned accept;
  float writeVal;
  float cx, cy, s0, s1;
};

// ---------------- workspace layout ----------------
#define OFF_EMB   ((size_t)1024)
#define OFF_SEED  (OFF_EMB  + (size_t)N_PIX * 8)
#define OFF_MASK  (OFF_SEED + (size_t)N_PIX * 4)
#define OFF_UNCL  (OFF_MASK + (size_t)NWORD * 4)
#define OFF_PROP  (OFF_UNCL + (size_t)NWORD * 4)

__device__ __forceinline__ void grid_sync(Ctl* c) {
  __syncthreads();
  if (threadIdx.x == 0) {
    __threadfence();
    unsigned gen = __atomic_load_n(&c->barGen, __ATOMIC_ACQUIRE);
    unsigned arrived = atomicAdd(&c->barCounter, 1u);
    if (arrived == (unsigned)gridDim.x - 1u) {
      c->barCounter = 0u;
      __threadfence();
      atomicAdd(&c->barGen, 1u);
    } else {
      while (__atomic_load_n(&c->barGen, __ATOMIC_ACQUIRE) == gen) {
        __builtin_amdgcn_s_sleep(2);
      }
    }
    __threadfence();
  }
  __syncthreads();
}

// exponent = s0*dx^2 + s1*dy^2 for 32 pixels/wave via two V_WMMA_F32_16X16X4_F32.
// A (16x4) rows all = [s0,s1,0,0]; B (4x16) col n = [dx2_n, dy2_n, 0, 0].
// D[m][n] = exponent(pixel n) for every m; D-VGPR0 lane l = exponent(pixel l%16).
__device__ __forceinline__ float wmma_expo(float dx2, float dy2, float s0, float s1,
                                           int lane) {
  bool lo = lane < 16;
  v2f a, b, b2;
  a[0] = lo ? s0 : 0.0f;  a[1] = lo ? s1 : 0.0f;       // A vgpr0: K=0 / K=2(=0)
  b[0] = lo ? dx2 : 0.0f; b[1] = lo ? dy2 : 0.0f;      // B rows K=0,K=1 (pixels 0..15)
  float dx2h = __shfl_xor(dx2, 16, 32);                // fixed half-wave swap
  float dy2h = __shfl_xor(dy2, 16, 32);
  b2[0] = lo ? dx2h : 0.0f; b2[1] = lo ? dy2h : 0.0f;
  v8f c0 = {}; v8f c1 = {};
  c0 = __builtin_amdgcn_wmma_f32_16x16x4_f32(false, a, false, b,  (short)0, c0, false, false);
  c1 = __builtin_amdgcn_wmma_f32_16x16x4_f32(false, a, false, b2, (short)0, c1, false, false);
  return lo ? c0[0] : c1[0];   // lane l<16 -> pixel l ; lane>=16 -> pixel 16+(l-16)=l
}

__global__ void init_ctl_kernel(Ctl* c) {
  c->barCounter = 0u; c->barGen = 0u;
  c->finished = 0u;
  c->argmaxPacked = 0ull;
  c->unclCnt = 0u; c->psum = 0u; c->overlap = 0u;
  c->count = 1u; c->accept = 0u;
  c->writeVal = 0.0f;
  c->cx = 0.0f; c->cy = 0.0f; c->s0 = 0.0f; c->s1 = 0.0f;
}

__global__ void __launch_bounds__(NTHR)
cluster_persistent(const float* __restrict__ pred, float* __restrict__ out,
                   unsigned char* __restrict__ ws) {
  Ctl* c = (Ctl*)ws;
  volatile Ctl* vc = (volatile Ctl*)c;
  float2*   emb  = (float2*)  (ws + OFF_EMB);
  float*    seed = (float*)   (ws + OFF_SEED);
  unsigned* mask = (unsigned*)(ws + OFF_MASK);
  unsigned* uncl = (unsigned*)(ws + OFF_UNCL);
  unsigned* prop = (unsigned*)(ws + OFF_PROP);

  const int lane = threadIdx.x & 31;
  const unsigned gw = (blockIdx.x * blockDim.x + threadIdx.x) >> 5;  // global wave id
  const bool master = (blockIdx.x == 0 && threadIdx.x == 0);

#if HAVE_ASYNC_LDS
  __shared__ float2 stage[2][NTHR];     // per-wave private double buffer (no x-wave use)
  const int slot = (int)threadIdx.x;    // wave w uses [w*32, w*32+32)
#endif

  // ---------------- Phase 0: precompute emb / seed / mask / uncl, zero output -------
  for (unsigned w = gw; w < NWORD; w += NWAVE) {
    unsigned i = w * 32u + (unsigned)lane;
    float p0 = pred[i];
    float p1 = pred[N_PIX + i];
    float p6 = pred[6u * N_PIX + i];
    unsigned col = i & (W_ - 1u);
    unsigned row = i >> 11;
    float x = (float)col * (2.0f / 2047.0f);
    float y = (float)row * (1.0f / 1023.0f);
    float2 e;
    e.x = tanhf(p0) + x;
    e.y = tanhf(p1) + y;
    float sv = 1.0f / (1.0f + __expf(-p6));
    emb[i]  = e;
    seed[i] = sv;
    unsigned mbits = (unsigned)__ballot(sv > MIN_SEED);
    if (lane == 0) { mask[w] = mbits; uncl[w] = mbits; }
    out[i] = 0.0f;
  }
  grid_sync(c);

  // ---------------- Sequential clustering loop (device-resident) --------------------
  for (int iter = 0; iter < MAXIT; ++iter) {
    // --- Step A: argmax(seed * uncl) + popcount(uncl) ---
    unsigned long long best = 0ull;
    unsigned localCnt = 0u;
    for (unsigned w = gw; w < NWORD; w += NWAVE) {
      unsigned uw = uncl[w];
      unsigned i  = w * 32u + (unsigned)lane;
      float sv = seed[i];
      bool active = (uw >> lane) & 1u;
      if (active) {
        unsigned long long pk =
            ((unsigned long long)__float_as_uint(sv) << 32) | (unsigned)(~i);
        if (pk > best) best = pk;
      }
      if (lane == 0) localCnt += (unsigned)__popc(uw);
    }
    for (int off = 16; off; off >>= 1) {
      unsigned long long o = __shfl_xor(best, off, 32);
      if (o > best) best = o;
    }
    if (lane == 0) {
      if (best) atomicMax(&c->argmaxPacked, best);
      if (localCnt) atomicAdd(&c->unclCnt, localCnt);
    }
    grid_sync(c);

    // --- Decision A (single thread) ---
    if (master) {
      unsigned long long pk = c->argmaxPacked;
      unsigned cnt = c->unclCnt;
      if (cnt <= (unsigned)MIN_PIXEL || pk == 0ull) {
        c->finished = 1u;
      } else {
        unsigned sidx  = ~((unsigned)pk);
        float    score = __uint_as_float((unsigned)(pk >> 32));
        if (score < THRESHOLD) {
          c->finished = 1u;     // go==false: body is a no-op then loop exits
        } else {
          float2 ce = emb[sidx];
          c->cx = ce.x;  c->cy = ce.y;
          c->s0 = __expf(pred[2u * N_PIX + sidx] * 10.0f);
          c->s1 = __expf(pred[3u * N_PIX + sidx] * 10.0f);
          uncl[sidx >> 5] &= ~(1u << (sidx & 31u));   // uncl2 = uncl \ {sidx}
          c->psum = 0u; c->overlap = 0u;
        }
      }
      c->argmaxPacked = 0ull;   // consumed; reset for next iteration
      c->unclCnt = 0u;
    }
    grid_sync(c);
    if (vc->finished) break;

    // --- Step B: proposal = (exp(-(s0*dx^2+s1*dy^2)) > 0.5) & mask ; sums ---
    const float cx = vc->cx, cy = vc->cy, ls0 = vc->s0, ls1 = vc->s1;
    unsigned lpsum = 0u, lovl = 0u;

#if HAVE_ASYNC_LDS
    // Prime buffer 0 with this wave's first word (async DMA, no VGPR staging).
    async_copy_b64(&emb[gw * 32u + (unsigned)lane], &stage[0][slot]);
    int buf = 0;
    for (unsigned w = gw; w < NWORD; w += NWAVE) {
      unsigned wn = w + NWAVE;
      if (wn < NWORD) {
        async_copy_b64(&emb[wn * 32u + (unsigned)lane], &stage[buf ^ 1][slot]);
        __builtin_amdgcn_s_wait_asynccnt(1);   // current buffer ready
      } else {
        __builtin_amdgcn_s_wait_asynccnt(0);   // last buffer ready
      }
      unsigned mw = mask[w];
      unsigned uw = uncl[w];
      float2 e = stage[buf][slot];
      float dx = e.x - cx, dy = e.y - cy;
      float expo = wmma_expo(dx * dx, dy * dy, ls0, ls1, lane);
      float dist = __expf(-expo);
      bool p = (dist > DIST_THRESH) && (((mw >> lane) & 1u) != 0u);
      unsigned pw = (unsigned)__ballot(p);
      if (lane == 0) {
        prop[w] = pw;
        lpsum += (unsigned)__popc(pw);
        lovl  += (unsigned)__popc(pw & uw);
      }
      buf ^= 1;
    }
#else
    for (unsigned w = gw; w < NWORD; w += NWAVE) {
      if (w + NWAVE < NWORD)
        __builtin_prefetch(&emb[(w + NWAVE) * 32u + (unsigned)lane], 0, 1);
      unsigned i  = w * 32u + (unsigned)lane;
      unsigned mw = mask[w];
      unsigned uw = uncl[w];
      float2 e = emb[i];
      float dx = e.x - cx, dy = e.y - cy;
      float expo = wmma_expo(dx * dx, dy * dy, ls0, ls1, lane);
      float dist = __expf(-expo);
      bool p = (dist > DIST_THRESH) && (((mw >> lane) & 1u) != 0u);
      unsigned pw = (unsigned)__ballot(p);
      if (lane == 0) {
        prop[w] = pw;
        lpsum += (unsigned)__popc(pw);
        lovl  += (unsigned)__popc(pw & uw);
      }
    }
#endif
    if (lane == 0) {
      if (lpsum) atomicAdd(&c->psum, lpsum);
      if (lovl)  atomicAdd(&c->overlap, lovl);
    }
    grid_sync(c);

    // --- Decision B ---
    if (master) {
      unsigned ps = c->psum, ov = c->overlap;
      float ratio = (float)ov / (float)(ps > 0u ? ps : 1u);
      unsigned acc = (ps > (unsigned)MIN_INST_PIXEL) && (ratio > INST_RATIO);
      c->accept = acc;
      if (acc) { c->writeVal = (float)c->count; c->count += 1u; }
    }
    grid_sync(c);

    // --- Step C: uncl = uncl2 & ~proposal ; labels where accepted ---
    const bool  acc = vc->accept != 0u;
    const float wv  = vc->writeVal;
    for (unsigned w = gw; w < NWORD; w += NWAVE) {
      unsigned pw = prop[w];
      unsigned uw = uncl[w];
      if (lane == 0) uncl[w] = uw & ~pw;
      if (acc && ((pw >> lane) & 1u))
        out[w * 32u + (unsigned)lane] = wv;
    }
    grid_sync(c);
  }
}

extern "C" void kernel_launch(void* const* d_in, const int* in_sizes, int n_in,
                              void* d_out, int out_size, void* d_ws, size_t ws_size,
                              hipStream_t stream) {
  (void)in_sizes; (void)n_in; (void)out_size; (void)ws_size;
  const float* pred = (const float*)d_in[0];
  float* out = (float*)d_out;
  unsigned char* ws = (unsigned char*)d_ws;
  init_ctl_kernel<<<1, 1, 0, stream>>>((Ctl*)ws);
  cluster_persistent<<<NBLK, NTHR, 0, stream>>>(pred, out, ws);
}